// QuantumNeuralLayer_51049981280812
// MI455X (gfx1250) — compile-verified
//
#include <hip/hip_runtime.h>
#include <math.h>

// ---------------------------------------------------------------------------
// MI455X (gfx1250) implementation.
//
// Algebraic fusion: both linear layers are linear, so
//   fused = X @ Weff^T + beff,
//   Weff[o,i] = sum_c Wf[o,c] * W16v[c,i]   (W16 [16,2048,2048] viewed [32768,2048])
//   beff[o]   = bf[o] + sum_c Wf[o,c] * b16[c]
// 2.2 TFLOP (naive) -> ~0.34 TFLOP and ~0.9 GB HBM traffic (weights streamed
// once): memory-bound at ~40us on 23.3 TB/s. Precision stays exact fp32 via
// V_WMMA_F32_16X16X4_F32. Data movement uses the CDNA5 async DMA path:
// GLOBAL_LOAD_ASYNC_TO_LDS_B128 + ASYNCcnt, double-buffered LDS, one barrier
// per K-slab.
// ---------------------------------------------------------------------------

typedef __attribute__((ext_vector_type(2))) float v2f;
typedef __attribute__((ext_vector_type(4))) float vf4;
typedef __attribute__((ext_vector_type(8))) float v8f;

#define DIM_I 2048
#define DIM_O 2048
#define DIM_C 32768   // 16 * 2048
#define ROWS  8192    // 4 * 2048

// Block tile: 256 threads = 8 wave32 waves in a 4x2 grid, each wave owns a
// 32x32 patch = 2x2 WMMA 16x16 tiles. BK=32 (8 wmma k-steps per slab).
#define BM 128
#define BN 64
#define BK 32
#define LDA_S  36               // padded stride (floats): 144B rows, 16B-aligned,
                                //   36*l mod 64 distinct for l=0..15 -> no conflicts
#define AS_BUF (BM * LDA_S)     // 4608 floats per buffer
#define LDB_NN 72               // k-major B tile: lane halves hit disjoint banks
#define LDB_NT 36               // n-major B tile (NT path), same pad as A
#define BS_BUF 2304             // floats per buffer (32*72 == 64*36)

// Memory -> LDS async DMA (tracked by ASYNCcnt). VDST is the wave-relative
// LDS byte offset == low 32 bits of a generic LDS pointer (ISA aperture map).
__device__ __forceinline__ void async_b128(const float* lds_ptr,
                                           const float* gptr) {
  uint32_t loff = (uint32_t)(uintptr_t)lds_ptr;
  asm volatile("global_load_async_to_lds_b128 %0, %1, off"
               :
               : "v"(loff), "v"(gptr)
               : "memory");
}
__device__ __forceinline__ void wait_async0() {
  asm volatile("s_wait_asynccnt 0x0" ::: "memory");
}

// D = A[M x K] * B + bias(row-broadcast over N), fp32 in/out.
// BT=false: B is [K x N] row-major (ldb = N stride)  -> NN GEMM
// BT=true : B is [N x K] row-major (ldb = K stride)  -> D = A * B^T
template <bool BT>
__global__ __launch_bounds__(256) void gemm_wmma_f32(
    const float* __restrict__ A, const float* __restrict__ B,
    const float* __restrict__ bias, float* __restrict__ D,
    int Kd, int lda, int ldb, int ldd) {
  __shared__ __align__(16) float As[2 * AS_BUF];
  __shared__ __align__(16) float Bs[2 * BS_BUF];

  const int t    = threadIdx.x;
  const int m0   = blockIdx.y * BM;
  const int n0   = blockIdx.x * BN;
  const int w    = t >> 5;          // wave id 0..7
  const int lane = t & 31;
  const int wm   = w & 3;           // wave row 0..3  (32 rows each)
  const int wn   = w >> 2;          // wave col 0..1  (32 cols each)
  const int half = lane >> 4;       // 0: lanes 0-15, 1: lanes 16-31
  const int lr   = lane & 15;

  v8f acc[2][2] = {};               // 4 x (16x16 f32) accumulators

  // Issue one K-slab of async memory->LDS copies into buffer `sb`.
  auto stage = [&](int sb, int kk0) {
    // A tile: BM x BK = 1024 float4, 4 per thread.
#pragma unroll
    for (int j = 0; j < 4; ++j) {
      const int f   = t * 4 + j;        // 0..1023
      const int row = f >> 3;
      const int kq  = (f & 7) << 2;
      async_b128(&As[sb * AS_BUF + row * LDA_S + kq],
                 &A[(size_t)(m0 + row) * lda + kk0 + kq]);
    }
    // B tile: 512 float4, 2 per thread.
#pragma unroll
    for (int j = 0; j < 2; ++j) {
      const int f = t * 2 + j;          // 0..511
      if constexpr (!BT) {              // k-major [BK][BN]
        const int kk = f >> 4;
        const int nq = (f & 15) << 2;
        async_b128(&Bs[sb * BS_BUF + kk * LDB_NN + nq],
                   &B[(size_t)(kk0 + kk) * ldb + n0 + nq]);
      } else {                          // n-major [BN][BK] (native layout)
        const int n  = f >> 3;
        const int kq = (f & 7) << 2;
        async_b128(&Bs[sb * BS_BUF + n * LDB_NT + kq],
                   &B[(size_t)(n0 + n) * ldb + kk0 + kq]);
      }
    }
  };

  stage(0, 0);
  int buf = 0;
  for (int k0 = 0; k0 < Kd; k0 += BK) {
    wait_async0();        // my slab's DMA done
    __syncthreads();      // everyone's DMA visible; prev slab fully consumed
    if (k0 + BK < Kd) stage(buf ^ 1, k0 + BK);
    if (k0 + 2 * BK < Kd)  // light prefetch of the slab after next
      __builtin_prefetch(&A[(size_t)(m0 + (t >> 1)) * lda + k0 + 2 * BK], 0, 1);

    const int ab = buf * AS_BUF;
    const int bb = buf * BS_BUF;
#pragma unroll
    for (int kt = 0; kt < BK / 4; ++kt) {
      const int kk4 = kt * 4 + half * 2;   // this lane-half's K pair
      // A fragment 16x4: lanes 0-15 hold K={0,1}, lanes 16-31 hold K={2,3}
      v2f af[2];
#pragma unroll
      for (int tm = 0; tm < 2; ++tm)
        af[tm] = *reinterpret_cast<const v2f*>(
            &As[ab + (wm * 32 + tm * 16 + lr) * LDA_S + kk4]);
      // B fragment 4x16: lane holds column N=lr; v0/v1 = two K rows,
      // lane half selects K+2.
      v2f bfr[2];
#pragma unroll
      for (int tn = 0; tn < 2; ++tn) {
        const int col = wn * 32 + tn * 16 + lr;
        if constexpr (!BT) {
          bfr[tn][0] = Bs[bb + (kk4 + 0) * LDB_NN + col];
          bfr[tn][1] = Bs[bb + (kk4 + 1) * LDB_NN + col];
        } else {                         // transpose falls out of n-major tile
          bfr[tn] = *reinterpret_cast<const v2f*>(
              &Bs[bb + col * LDB_NT + kk4]);
        }
      }
#pragma unroll
      for (int tm = 0; tm < 2; ++tm)
#pragma unroll
        for (int tn = 0; tn < 2; ++tn)
          acc[tm][tn] = __builtin_amdgcn_wmma_f32_16x16x4_f32(
              false, af[tm], false, bfr[tn], (short)0, acc[tm][tn],
              false, false);
    }
    buf ^= 1;
  }

  // Epilogue: C/D layout: VGPR j -> M=j (lanes 0-15) / M=8+j (lanes 16-31).
#pragma unroll
  for (int tm = 0; tm < 2; ++tm) {
    const int gmb = m0 + wm * 32 + tm * 16 + half * 8;
#pragma unroll
    for (int tn = 0; tn < 2; ++tn) {
      const int gn = n0 + wn * 32 + tn * 16 + lr;
      const float bv = bias ? bias[gn] : 0.0f;
#pragma unroll
      for (int j = 0; j < 8; ++j)
        D[(size_t)(gmb + j) * ldd + gn] = acc[tm][tn][j] + bv;
    }
  }
}

// beff[o] = bf[o] + sum_c Wf[o,c] * b16_flat[c]
__global__ __launch_bounds__(256) void beff_kernel(
    const float* __restrict__ Wf, const float* __restrict__ b16,
    const float* __restrict__ bf, float* __restrict__ beff) {
  __shared__ float red[256];
  const int o = blockIdx.x, t = threadIdx.x;
  float s = 0.f;
  for (int c = t; c < DIM_C; c += 256)
    s += Wf[(size_t)o * DIM_C + c] * b16[c];
  red[t] = s;
  __syncthreads();
  for (int off = 128; off > 0; off >>= 1) {
    if (t < off) red[t] += red[t + off];
    __syncthreads();
  }
  if (t == 0) beff[o] = red[0] + bf[o];
}

// In-place LayerNorm(eps=1e-5) + exact erf-GELU over rows of 2048 f32.
__global__ __launch_bounds__(256) void ln_gelu_kernel(
    float* __restrict__ out, const float* __restrict__ gamma,
    const float* __restrict__ beta) {
  __shared__ float s1[256], s2[256];
  const int r = blockIdx.x, t = threadIdx.x;
  float* row = out + (size_t)r * DIM_O;

  float x[8], sum = 0.f, sq = 0.f;
  vf4 v0 = *reinterpret_cast<const vf4*>(&row[t * 8 + 0]);
  vf4 v1 = *reinterpret_cast<const vf4*>(&row[t * 8 + 4]);
#pragma unroll
  for (int j = 0; j < 4; ++j) { x[j] = v0[j]; x[4 + j] = v1[j]; }
#pragma unroll
  for (int j = 0; j < 8; ++j) { sum += x[j]; sq += x[j] * x[j]; }
  s1[t] = sum; s2[t] = sq;
  __syncthreads();
  for (int off = 128; off > 0; off >>= 1) {
    if (t < off) { s1[t] += s1[t + off]; s2[t] += s2[t + off]; }
    __syncthreads();
  }
  const float mean = s1[0] * (1.0f / DIM_O);
  const float var  = s2[0] * (1.0f / DIM_O) - mean * mean;
  const float inv  = rsqrtf(var + 1e-5f);

  vf4 o0, o1;
#pragma unroll
  for (int j = 0; j < 8; ++j) {
    const int c = t * 8 + j;
    const float g = (x[j] - mean) * inv * gamma[c] + beta[c];
    const float y = 0.5f * g * (1.0f + erff(g * 0.70710678118654752f));
    if (j < 4) o0[j] = y; else o1[j - 4] = y;
  }
  *reinterpret_cast<vf4*>(&row[t * 8 + 0]) = o0;
  *reinterpret_cast<vf4*>(&row[t * 8 + 4]) = o1;
}

extern "C" void kernel_launch(void* const* d_in, const int* in_sizes, int n_in,
                              void* d_out, int out_size, void* d_ws,
                              size_t ws_size, hipStream_t stream) {
  const float* x     = (const float*)d_in[0];  // [4,2048,2048]
  const float* W16   = (const float*)d_in[1];  // [16,2048,2048] == [32768,2048]
  const float* b16   = (const float*)d_in[2];  // [16,2048] == [32768]
  const float* Wf    = (const float*)d_in[3];  // [2048,32768]
  const float* bf    = (const float*)d_in[4];  // [2048]
  const float* gamma = (const float*)d_in[5];  // [2048]
  const float* beta  = (const float*)d_in[6];  // [2048]
  float* out  = (float*)d_out;                 // [8192,2048]
  float* Weff = (float*)d_ws;                  // 2048*2048 f32 = 16 MB
  float* beff = Weff + (size_t)DIM_O * DIM_I;  // 2048 f32

  dim3 blk(256);
  // 1) Weff = Wf [2048 x 32768] @ W16v [32768 x 2048]   (NN GEMM)
  gemm_wmma_f32<false><<<dim3(DIM_I / BN, DIM_O / BM), blk, 0, stream>>>(
      Wf, W16, nullptr, Weff, DIM_C, DIM_C, DIM_I, DIM_I);
  // 2) beff = bf + Wf @ b16_flat
  beff_kernel<<<DIM_O, blk, 0, stream>>>(Wf, b16, bf, beff);
  // 3) fused = X [8192 x 2048] @ Weff^T + beff          (NT GEMM)
  gemm_wmma_f32<true><<<dim3(DIM_O / BN, ROWS / BM), blk, 0, stream>>>(
      x, Weff, beff, out, DIM_I, DIM_I, DIM_I, DIM_O);
  // 4) LayerNorm + exact GELU, in place on d_out
  ln_gelu_kernel<<<ROWS, blk, 0, stream>>>(out, gamma, beta);
}